// SO3_Linear_22419729285494
// MI455X (gfx1250) — compile-verified
//
#include <hip/hip_runtime.h>

typedef __attribute__((ext_vector_type(2))) float v2f;
typedef __attribute__((ext_vector_type(8))) float v8f;

namespace {
constexpr int kRows    = 16;          // batch rows per workgroup
constexpr int kFlat    = 512;         // MUL * (1+3+5+7)
constexpr int kXStride = kFlat + 4;   // 516 floats: 16B-aligned, de-conflicts banks
constexpr int kWStride = 33;          // padded W row (conflict-free A reads)
constexpr int kWSeg    = 32 * kWStride;
}

__global__ __launch_bounds__(256)
void so3_linear_wmma_f32(const float* __restrict__ x,
                         const float* __restrict__ W0,
                         const float* __restrict__ W1,
                         const float* __restrict__ W2,
                         const float* __restrict__ W3,
                         const float* __restrict__ bias,
                         float* __restrict__ out) {
  __shared__ float lds_x[kRows * kXStride];   // 33,024 B: x tile, reused as y tile
  __shared__ float lds_w[4 * kWSeg];          // 16,896 B: padded weights

  const int tid = threadIdx.x;
  const size_t base = (size_t)blockIdx.x * (size_t)(kRows * kFlat);

  // ---- stage the 4 weight matrices into padded LDS rows ----
  for (int i = tid; i < 4 * 1024; i += 256) {
    const int l = i >> 10;
    const int e = i & 1023;
    const float* w = (l == 0) ? W0 : (l == 1) ? W1 : (l == 2) ? W2 : W3;
    lds_w[l * kWSeg + (e >> 5) * kWStride + (e & 31)] = w[e];
  }
  // ---- stage 16 batch rows of x: coalesced b128 global loads ----
  for (int i = tid; i < (kRows * kFlat) / 4; i += 256) {
    const int g   = i * 4;
    const int row = g >> 9;          // g / 512
    const int col = g & (kFlat - 1); // g % 512
    const float4 v = *(const float4*)(x + base + g);
    *(float4*)(&lds_x[row * kXStride + col]) = v;
  }
  __syncthreads();

  const int wave  = tid >> 5;
  const int lane  = tid & 31;
  const int lhalf = lane >> 4;   // K-half for A/B operands, M-half for C/D
  const int lrow  = lane & 15;   // A: row M; B: column N (= local batch row)

  v8f acc[4];
  int tl[4], tj[4], th[4], td[4], tstart[4];

  // 32 output tiles per block (2 halves x 16 j-slots); 4 tiles per wave.
#pragma unroll
  for (int ti = 0; ti < 4; ++ti) {
    const int t    = wave * 4 + ti;
    const int half = t & 1;
    const int t2   = t >> 1;          // j-slot 0..15 across irreps (cum d: 0,1,4,9,16)
    int l, j, d, start;
    if (t2 < 1)      { l = 0; j = t2;     d = 1; start = 0;   }
    else if (t2 < 4) { l = 1; j = t2 - 1; d = 3; start = 32;  }
    else if (t2 < 9) { l = 2; j = t2 - 4; d = 5; start = 128; }
    else             { l = 3; j = t2 - 9; d = 7; start = 288; }
    tl[ti] = l; tj[ti] = j; th[ti] = half; td[ti] = d; tstart[ti] = start;

    const int n0    = half * 16;
    const int abase = l * kWSeg + (n0 + lrow) * kWStride + 2 * lhalf;
    const int bbase = lrow * kXStride + start + j + 2 * lhalf * d;

    v8f c = {0.f, 0.f, 0.f, 0.f, 0.f, 0.f, 0.f, 0.f};
#pragma unroll
    for (int k = 0; k < 8; ++k) {     // K=32 as 8 chained K=4 WMMAs
      v2f a, b;
      a.x = lds_w[abase + 4 * k];             // W[n0+lrow, 4k+2*lhalf]
      a.y = lds_w[abase + 4 * k + 1];         // W[n0+lrow, 4k+2*lhalf+1]
      b.x = lds_x[bbase + (4 * k) * d];       // X[b=lrow, m=4k+2*lhalf, j]
      b.y = lds_x[bbase + (4 * k) * d + d];   // X[b=lrow, m+1, j]
      c = __builtin_amdgcn_wmma_f32_16x16x4_f32(false, a, false, b,
                                                (short)0, c, false, false);
    }
    acc[ti] = c;
  }

  __syncthreads();   // all reads of lds_x done; reuse it as the y staging tile

#pragma unroll
  for (int ti = 0; ti < 4; ++ti) {
    const int n0    = th[ti] * 16 + 8 * lhalf;  // D layout: M = v + 8*(lane/16)
    const int d     = td[ti];
    const int start = tstart[ti];
    const int j     = tj[ti];
#pragma unroll
    for (int v = 0; v < 8; ++v) {
      const int n = n0 + v;
      float val = acc[ti][v];
      if (tl[ti] == 0) val += bias[n];          // bias only on l=0 segment
      lds_x[lrow * kXStride + start + n * d + j] = val;
    }
  }
  __syncthreads();

  // ---- coalesced b128 stores of the y tile ----
  for (int i = tid; i < (kRows * kFlat) / 4; i += 256) {
    const int g   = i * 4;
    const int row = g >> 9;
    const int col = g & (kFlat - 1);
    *(float4*)(out + base + g) = *(const float4*)(&lds_x[row * kXStride + col]);
  }
}

extern "C" void kernel_launch(void* const* d_in, const int* in_sizes, int n_in,
                              void* d_out, int out_size, void* d_ws, size_t ws_size,
                              hipStream_t stream) {
  const float* x    = (const float*)d_in[0];
  const float* W0   = (const float*)d_in[1];
  const float* W1   = (const float*)d_in[2];
  const float* W2   = (const float*)d_in[3];
  const float* W3   = (const float*)d_in[4];
  const float* bias = (const float*)d_in[5];
  float* out = (float*)d_out;

  const int Bn = in_sizes[0] / kFlat;   // 131072
  const int grid = Bn / kRows;          // 8192 workgroups x 16 rows
  so3_linear_wmma_f32<<<grid, 256, 0, stream>>>(x, W0, W1, W2, W3, bias, out);
}